// QuantumLayer_90752658965227
// MI455X (gfx1250) — compile-verified
//
#include <hip/hip_runtime.h>

// -------- CDNA5 (gfx1250) batched 12-qubit statevector simulator --------
// One block = one batch sample; state (re/im SoA) lives in LDS.
// Per layer, valid reordering of the reference gate sequence:
//   A1 -> A2 -> [P23*P12*P01*A0] -> P34 -> [P67*P56*P45] -> P78
//      -> [P(10,11)*P(9,10)*P89] -> P(11,0)
// where Ag = kron of fused RZ*RY*RX over a 4-qubit group and bracketed ops
// are 16x16 complex unitaries applied with v_wmma_f32_16x16x4_f32.
// Cross-group pairs (3,4),(7,8),(11,0) fused CRZ*XX*YY applied in VALU.

typedef float v2f __attribute__((ext_vector_type(2)));
typedef float v8f __attribute__((ext_vector_type(8)));

struct c32 { float x, y; };
__device__ __forceinline__ c32 cmul(c32 a, c32 b) {
    return { a.x * b.x - a.y * b.y, a.x * b.y + a.y * b.x };
}

#define NQ     12
#define DIM    4096     // 2^12
#define NLAYER 4

__global__ __launch_bounds__(256)
void qsim_kernel(const float* __restrict__ xin,
                 const float* __restrict__ params,
                 float* __restrict__ out) {
    __shared__ float sre[DIM];
    __shared__ float sim_[DIM];
    __shared__ float bufAr[256], bufAi[256];   // 16x16 unitary ping
    __shared__ float bufBr[256], bufBi[256];   // 16x16 unitary pong
    __shared__ float uq[NQ][8];     // per-qubit fused 2x2 (4 complex)
    __shared__ float pv[NQ][6];     // per-pair entangler scalars (cross pairs)
    __shared__ float pmr[NQ][16];   // per-pair fused 4x4 (real)
    __shared__ float pmi[NQ][16];   // per-pair fused 4x4 (imag)
    __shared__ float enc[NQ][4];    // encoding per-qubit 2-vector
    __shared__ float red[256 * NQ]; // reduction buffer

    const int tid  = threadIdx.x;
    const int b    = blockIdx.x;
    const int lane = tid & 31;
    const int wv   = tid >> 5;        // wave id, 0..7
    const int half = lane >> 4;       // 0: lanes 0-15, 1: lanes 16-31
    const int lr   = lane & 15;

    // Apply 16x16 complex unitary W to state viewed as 16x256 for group g.
    auto wmma_apply = [&](const float* Wr, const float* Wi, int g) {
        const int rstride = (g == 0) ? 256 : ((g == 1) ? 16 : 1);
        const int nstride = (g == 2) ? 16 : 1;
        for (int tt = 0; tt < 2; ++tt) {
            int ct = 2 * wv + tt;                    // column tile 0..15
            int tbase = (g == 0) ? ct * 16 : ct * 256;
            v8f accR = {}; v8f accI = {};
#pragma unroll
            for (int ks = 0; ks < 4; ++ks) {
                int kk0 = 4 * ks + 2 * half;
                v2f ar, ai, ain, br, bi;
                ar[0] = Wr[lr * 16 + kk0];  ar[1] = Wr[lr * 16 + kk0 + 1];
                ai[0] = Wi[lr * 16 + kk0];  ai[1] = Wi[lr * 16 + kk0 + 1];
                ain = -ai;
                int i0 = tbase + kk0 * rstride + lr * nstride;
                br[0] = sre[i0];  br[1] = sre[i0 + rstride];
                bi[0] = sim_[i0]; bi[1] = sim_[i0 + rstride];
                accR = __builtin_amdgcn_wmma_f32_16x16x4_f32(
                    false, ar, false, br, (short)0, accR, false, false);
                accR = __builtin_amdgcn_wmma_f32_16x16x4_f32(
                    false, ain, false, bi, (short)0, accR, false, false);
                accI = __builtin_amdgcn_wmma_f32_16x16x4_f32(
                    false, ar, false, bi, (short)0, accI, false, false);
                accI = __builtin_amdgcn_wmma_f32_16x16x4_f32(
                    false, ai, false, br, (short)0, accI, false, false);
            }
#pragma unroll
            for (int j = 0; j < 8; ++j) {
                int M = j + 8 * half;
                int o = tbase + M * rstride + lr * nstride;
                sre[o] = accR[j]; sim_[o] = accI[j];
            }
        }
    };

    // Build kron(u[4g],u[4g+1],u[4g+2],u[4g+3]) into out buffers.
    auto kron_build = [&](int g, float* outR, float* outI) {
        int r = tid >> 4, c = tid & 15, q0 = 4 * g;
        int o0 = ((((r >> 3) & 1) * 2 + ((c >> 3) & 1)) * 2);
        int o1 = ((((r >> 2) & 1) * 2 + ((c >> 2) & 1)) * 2);
        int o2 = ((((r >> 1) & 1) * 2 + ((c >> 1) & 1)) * 2);
        int o3 = (((r & 1) * 2 + (c & 1)) * 2);
        c32 acc = { uq[q0][o0], uq[q0][o0 + 1] };
        acc = cmul(acc, c32{ uq[q0 + 1][o1], uq[q0 + 1][o1 + 1] });
        acc = cmul(acc, c32{ uq[q0 + 2][o2], uq[q0 + 2][o2 + 1] });
        acc = cmul(acc, c32{ uq[q0 + 3][o3], uq[q0 + 3][o3 + 1] });
        outR[tid] = acc.x; outI[tid] = acc.y;
    };

    // out = (P_pair (x) I) * in, pair acts on group bits (3-t, 2-t).
    auto chain_step = [&](int pair, int t,
                          const float* inR, const float* inI,
                          float* outR, float* outI) {
        int r = tid >> 4, c = tid & 15;
        int sh = 2 - t;
        int rowsel = ((r >> sh) & 3) * 4;
        int rmask = r & ~(3 << sh);
        c32 acc = { 0.0f, 0.0f };
#pragma unroll
        for (int j = 0; j < 4; ++j) {
            int k = rmask | (j << sh);
            c32 p = { pmr[pair][rowsel + j], pmi[pair][rowsel + j] };
            c32 v = { inR[k * 16 + c], inI[k * 16 + c] };
            acc.x += p.x * v.x - p.y * v.y;
            acc.y += p.x * v.y + p.y * v.x;
        }
        outR[tid] = acc.x; outI[tid] = acc.y;
    };

    // Direct build of (P_pair (x) I) on bits (3,2)  [t = 0].
    auto pfull_build = [&](int pair, float* outR, float* outI) {
        int r = tid >> 4, c = tid & 15;
        if ((r & 3) == (c & 3)) {
            int idx = ((r >> 2) & 3) * 4 + ((c >> 2) & 3);
            outR[tid] = pmr[pair][idx]; outI[tid] = pmi[pair][idx];
        } else {
            outR[tid] = 0.0f; outI[tid] = 0.0f;
        }
    };

    // Cross-group fused CRZ*XX*YY pair applied directly on amplitude quads.
    auto pair_sweep = [&](int i) {
        int j  = (i + 1) % NQ;
        int pi_ = 11 - i, pj_ = 11 - j;
        int lo = pi_ < pj_ ? pi_ : pj_;
        int hi = pi_ < pj_ ? pj_ : pi_;
        float cd = pv[i][0], sd = pv[i][1];
        float cs = pv[i][2], ss = pv[i][3];
        float cph = pv[i][4], sph = pv[i][5];
#pragma unroll
        for (int m = 0; m < 4; ++m) {
            int qd   = tid + 256 * m;
            int low  = qd & ((1 << lo) - 1);
            int mid  = (qd >> lo) & ((1 << (hi - 1 - lo)) - 1);
            int high = qd >> (hi - 1);
            int base = (high << (hi + 1)) | (mid << (lo + 1)) | low;
            int i01 = base | (1 << pj_);
            int i10 = base | (1 << pi_);
            int i11 = i10  | (1 << pj_);
            c32 a00 = { sre[base], sim_[base] };
            c32 a01 = { sre[i01],  sim_[i01]  };
            c32 a10 = { sre[i10],  sim_[i10]  };
            c32 a11 = { sre[i11],  sim_[i11]  };
            c32 b11 = { cph * a11.x - sph * a11.y, cph * a11.y + sph * a11.x };
            c32 b10 = { cph * a10.x + sph * a10.y, cph * a10.y - sph * a10.x };
            c32 n00 = {  cd * a00.x + sd * b11.y,  cd * a00.y - sd * b11.x };
            c32 n11 = {  sd * a00.y + cd * b11.x, -sd * a00.x + cd * b11.y };
            c32 n01 = {  cs * a01.x + ss * b10.y,  cs * a01.y - ss * b10.x };
            c32 n10 = {  ss * a01.y + cs * b10.x, -ss * a01.x + cs * b10.y };
            sre[base] = n00.x; sim_[base] = n00.y;
            sre[i01]  = n01.x; sim_[i01]  = n01.y;
            sre[i10]  = n10.x; sim_[i10]  = n10.y;
            sre[i11]  = n11.x; sim_[i11]  = n11.y;
        }
    };

    // ---- encoding factors: RY(x)*RZ(x^2) applied to |0> per wire ----
    if (tid < NQ) {
        float xv = xin[b * NQ + tid];
        float h = 0.5f * xv;
        float a = 0.5f * xv * xv;
        float ch = cosf(h), sh = sinf(h);
        float ca = cosf(a), sa = sinf(a);
        enc[tid][0] = ch * ca;  enc[tid][1] = -ch * sa;
        enc[tid][2] = sh * ca;  enc[tid][3] =  sh * sa;
    }
    __syncthreads();

    // ---- initial product state ----
    for (int m = 0; m < 16; ++m) {
        int k = tid + 256 * m;
        c32 amp = { 1.0f, 0.0f };
#pragma unroll
        for (int q = 0; q < NQ; ++q) {
            int bit = (k >> (11 - q)) & 1;
            c32 f = { enc[q][bit * 2], enc[q][bit * 2 + 1] };
            amp = cmul(amp, f);
        }
        sre[k] = amp.x; sim_[k] = amp.y;
    }
    __syncthreads();

    for (int l = 0; l < NLAYER; ++l) {
        // ---- per-layer scalar precompute (uniform across block) ----
        if (tid < NQ) {
            const float* wp = params + (l * NQ + tid) * 3;
            float w0 = wp[0], w1 = wp[1], w2 = wp[2];
            float c0 = cosf(0.5f * w0), s0 = sinf(0.5f * w0);
            float c1 = cosf(0.5f * w1), s1 = sinf(0.5f * w1);
            float cz = cosf(0.5f * w2), sz = sinf(0.5f * w2);
            // u = RZ(w2)*RY(w1)*RX(w0)
            c32 A  = {  c1 * c0,  s1 * s0 };
            c32 Bm = { -s1 * c0, -c1 * s0 };
            c32 Cm = {  s1 * c0, -c1 * s0 };
            c32 Dm = {  c1 * c0, -s1 * s0 };
            c32 ez  = { cz,  sz };
            c32 ezc = { cz, -sz };
            c32 u00 = cmul(A, ezc), u01 = cmul(Bm, ezc);
            c32 u10 = cmul(Cm, ez), u11 = cmul(Dm, ez);
            uq[tid][0] = u00.x; uq[tid][1] = u00.y;
            uq[tid][2] = u01.x; uq[tid][3] = u01.y;
            uq[tid][4] = u10.x; uq[tid][5] = u10.y;
            uq[tid][6] = u11.x; uq[tid][7] = u11.y;
            // fused CRZ(w0)*XX(w1)*YY(w2)
            float cd = cosf(0.5f * (w1 - w2)), sd = sinf(0.5f * (w1 - w2));
            float cs = cosf(0.5f * (w1 + w2)), ss = sinf(0.5f * (w1 + w2));
            float cph = cosf(0.5f * w0), sph = sinf(0.5f * w0);
            pv[tid][0] = cd; pv[tid][1] = sd;
            pv[tid][2] = cs; pv[tid][3] = ss;
            pv[tid][4] = cph; pv[tid][5] = sph;
            // full 4x4 pair matrix (basis 2*bit_i + bit_j)
#pragma unroll
            for (int e = 0; e < 16; ++e) { pmr[tid][e] = 0.0f; pmi[tid][e] = 0.0f; }
            pmr[tid][0]  = cd;
            pmr[tid][3]  = sd * sph;  pmi[tid][3]  = -sd * cph;  // (-i sd)e^{+i phi}
            pmi[tid][12] = -sd;                                  // -i sd
            pmr[tid][15] = cd * cph;  pmi[tid][15] =  cd * sph;  // cd e^{+i phi}
            pmr[tid][5]  = cs;
            pmr[tid][6]  = -ss * sph; pmi[tid][6]  = -ss * cph;  // (-i ss)e^{-i phi}
            pmi[tid][9]  = -ss;                                  // -i ss
            pmr[tid][10] = cs * cph;  pmi[tid][10] = -cs * sph;  // cs e^{-i phi}
        }
        __syncthreads();

        // ---- A1, A2: kron 1q unitaries on groups 1, 2 ----
        kron_build(1, bufAr, bufAi);                 __syncthreads();
        wmma_apply(bufAr, bufAi, 1);                 __syncthreads();
        kron_build(2, bufAr, bufAi);                 __syncthreads();
        wmma_apply(bufAr, bufAi, 2);                 __syncthreads();

        // ---- G0 = P23*P12*P01*A0 on group 0 ----
        kron_build(0, bufAr, bufAi);                 __syncthreads();
        chain_step(0, 0, bufAr, bufAi, bufBr, bufBi); __syncthreads();
        chain_step(1, 1, bufBr, bufBi, bufAr, bufAi); __syncthreads();
        chain_step(2, 2, bufAr, bufAi, bufBr, bufBi); __syncthreads();
        wmma_apply(bufBr, bufBi, 0);                 __syncthreads();

        // ---- cross pair (3,4) ----
        pair_sweep(3);                               __syncthreads();

        // ---- W1 = P67*P56*P45 on group 1 ----
        pfull_build(4, bufAr, bufAi);                __syncthreads();
        chain_step(5, 1, bufAr, bufAi, bufBr, bufBi); __syncthreads();
        chain_step(6, 2, bufBr, bufBi, bufAr, bufAi); __syncthreads();
        wmma_apply(bufAr, bufAi, 1);                 __syncthreads();

        // ---- cross pair (7,8) ----
        pair_sweep(7);                               __syncthreads();

        // ---- W2 = P(10,11)*P(9,10)*P89 on group 2 ----
        pfull_build(8, bufAr, bufAi);                __syncthreads();
        chain_step(9, 1, bufAr, bufAi, bufBr, bufBi); __syncthreads();
        chain_step(10, 2, bufBr, bufBi, bufAr, bufAi); __syncthreads();
        wmma_apply(bufAr, bufAi, 2);                 __syncthreads();

        // ---- cross pair (11,0) ----
        pair_sweep(11);                              __syncthreads();
    }

    // ---- <Z_q> marginals ----
    float part[NQ];
#pragma unroll
    for (int q = 0; q < NQ; ++q) part[q] = 0.0f;
    for (int m = 0; m < 16; ++m) {
        int k = tid + 256 * m;
        float pr = sre[k] * sre[k] + sim_[k] * sim_[k];
#pragma unroll
        for (int q = 0; q < NQ; ++q)
            part[q] += ((k >> (11 - q)) & 1) ? -pr : pr;
    }
#pragma unroll
    for (int q = 0; q < NQ; ++q) red[tid * NQ + q] = part[q];
    __syncthreads();
    for (int s = 128; s > 0; s >>= 1) {
        if (tid < s) {
#pragma unroll
            for (int q = 0; q < NQ; ++q)
                red[tid * NQ + q] += red[(tid + s) * NQ + q];
        }
        __syncthreads();
    }
    if (tid < NQ) out[b * NQ + tid] = red[tid];
}

extern "C" void kernel_launch(void* const* d_in, const int* in_sizes, int n_in,
                              void* d_out, int out_size, void* d_ws, size_t ws_size,
                              hipStream_t stream) {
    (void)n_in; (void)out_size; (void)d_ws; (void)ws_size;
    const float* x      = (const float*)d_in[0];   // (512, 12) f32
    const float* params = (const float*)d_in[1];   // (144,)   f32
    float* out = (float*)d_out;                    // (512, 12) f32
    int B = in_sizes[0] / NQ;
    qsim_kernel<<<dim3(B), dim3(256), 0, stream>>>(x, params, out);
}